// Group_17678085390961
// MI455X (gfx1250) — compile-verified
//
#include <hip/hip_runtime.h>
#include <stdint.h>

#ifndef __has_builtin
#define __has_builtin(x) 0
#endif

#define B       8
#define NPTS    65536
#define NGROUP  512
#define GSIZE   64
#define WALKS   (B * NGROUP)
#define LDSP    16384                 // points staged in LDS: 16384*12B = 192 KB
#define FPS_THREADS 1024
#define PPT     (NPTS / FPS_THREADS)  // 64 points per thread
#define PPT_LDS (LDSP / FPS_THREADS)  // 16 of them LDS-resident

#if defined(__HIP_DEVICE_COMPILE__) && defined(__gfx1250__) && \
    __has_builtin(__builtin_amdgcn_tensor_load_to_lds) &&      \
    __has_builtin(__builtin_amdgcn_s_wait_tensorcnt)
#define USE_TDM 1
#else
#define USE_TDM 0
#endif

typedef unsigned int v4u __attribute__((ext_vector_type(4)));
typedef int          v8i __attribute__((ext_vector_type(8)));
typedef int          v4i __attribute__((ext_vector_type(4)));

// ---------------- threefry2x32 (bit-exact JAX original impl) ----------------
__device__ __forceinline__ uint32_t rotl32(uint32_t x, int r) {
    return (x << r) | (x >> (32 - r));
}
__device__ __forceinline__ void tf2x32(uint32_t k0, uint32_t k1,
                                       uint32_t x0, uint32_t x1,
                                       uint32_t& o0, uint32_t& o1) {
    uint32_t ks2 = 0x1BD11BDAu ^ k0 ^ k1;
    x0 += k0; x1 += k1;
    x0 += x1; x1 = rotl32(x1, 13); x1 ^= x0;
    x0 += x1; x1 = rotl32(x1, 15); x1 ^= x0;
    x0 += x1; x1 = rotl32(x1, 26); x1 ^= x0;
    x0 += x1; x1 = rotl32(x1,  6); x1 ^= x0;
    x0 += k1; x1 += ks2 + 1u;
    x0 += x1; x1 = rotl32(x1, 17); x1 ^= x0;
    x0 += x1; x1 = rotl32(x1, 29); x1 ^= x0;
    x0 += x1; x1 = rotl32(x1, 16); x1 ^= x0;
    x0 += x1; x1 = rotl32(x1, 24); x1 ^= x0;
    x0 += ks2; x1 += k0 + 2u;
    x0 += x1; x1 = rotl32(x1, 13); x1 ^= x0;
    x0 += x1; x1 = rotl32(x1, 15); x1 ^= x0;
    x0 += x1; x1 = rotl32(x1, 26); x1 ^= x0;
    x0 += x1; x1 = rotl32(x1,  6); x1 ^= x0;
    x0 += k0; x1 += k1 + 3u;
    x0 += x1; x1 = rotl32(x1, 17); x1 ^= x0;
    x0 += x1; x1 = rotl32(x1, 29); x1 ^= x0;
    x0 += x1; x1 = rotl32(x1, 16); x1 ^= x0;
    x0 += x1; x1 = rotl32(x1, 24); x1 ^= x0;
    x0 += k1; x1 += ks2 + 4u;
    x0 += x1; x1 = rotl32(x1, 13); x1 ^= x0;
    x0 += x1; x1 = rotl32(x1, 15); x1 ^= x0;
    x0 += x1; x1 = rotl32(x1, 26); x1 ^= x0;
    x0 += x1; x1 = rotl32(x1,  6); x1 ^= x0;
    o0 = x0 + ks2; o1 = x1 + k0 + 5u;
}

// ------------------------------ Kernel 1: FPS -------------------------------
__global__ __launch_bounds__(FPS_THREADS) void fps_kernel(
    const float* __restrict__ xyz, int* __restrict__ cidx,
    float* __restrict__ centerOut) {
    __shared__ float sxyz[LDSP * 3];   // 192 KB staged tile of this batch's xyz
    __shared__ float redD[32];
    __shared__ int   redI[32];
    __shared__ float cPt[3];

    const int b = blockIdx.x;
    const int t = threadIdx.x;
    const float* gx = xyz + (size_t)b * NPTS * 3;

#if USE_TDM
    if (t < 32) {  // one wave issues one TDM op; EXEC ignored by TDM
        uint64_t ga = (uint64_t)(uintptr_t)gx;
        uint32_t lo = (uint32_t)(uintptr_t)(&sxyz[0]);  // low 32b = LDS offset
        v4u g0;
        g0[0] = 1u;                                   // count=1, flags=0
        g0[1] = lo;                                   // lds_addr
        g0[2] = (uint32_t)ga;                         // global_addr[31:0]
        g0[3] = ((uint32_t)(ga >> 32) & 0x01FFFFFFu) | (2u << 30); // type=2
        const uint32_t dim0 = 12288u, dim1 = (uint32_t)(LDSP * 3 / 12288); // 4
        v8i g1;
        g1[0] = (int)(2u << 16);                      // data_size=4B, mask=0
        g1[1] = (int)((dim0 & 0xFFFFu) << 16);        // tensor_dim0 lo16
        g1[2] = (int)((dim0 >> 16) | ((dim1 & 0xFFFFu) << 16));
        g1[3] = (int)((dim1 >> 16) | (12288u << 16)); // tile_dim0 = 12288
        g1[4] = (int)dim1;                            // tile_dim1=4, tile_dim2=0
        g1[5] = (int)12288;                           // tensor_dim0_stride lo32
        g1[6] = 0;
        g1[7] = 0;
        v4i gz4  = {0, 0, 0, 0};
        v8i gz8  = {0, 0, 0, 0, 0, 0, 0, 0};
        __builtin_amdgcn_tensor_load_to_lds(g0, g1, gz4, gz4, gz8, 0);
        __builtin_amdgcn_s_wait_tensorcnt(0);
    }
    // Prefetch the non-LDS-resident tail into L2 (one byte per 128B line).
    for (int i = LDSP * 3 + t * 32; i < NPTS * 3; i += FPS_THREADS * 32)
        __builtin_prefetch(&gx[i], 0, 1);
#else
    for (int i = t; i < LDSP * 3; i += FPS_THREADS) sxyz[i] = gx[i];
#endif
    __syncthreads();

    if (t == 0) {
        cPt[0] = sxyz[0]; cPt[1] = sxyz[1]; cPt[2] = sxyz[2];
        cidx[b * NGROUP] = 0;
        float* co = centerOut + (size_t)b * NGROUP * 3;
        co[0] = sxyz[0]; co[1] = sxyz[1]; co[2] = sxyz[2];
    }
    __syncthreads();

    float dist[PPT];
#pragma unroll
    for (int j = 0; j < PPT; ++j) dist[j] = 1e10f;

    for (int it = 1; it < NGROUP; ++it) {
        const float px = cPt[0], py = cPt[1], pz = cPt[2];
        float bD = -1.0f; int bI = 0x7FFFFFFF;
#pragma unroll
        for (int j = 0; j < PPT_LDS; ++j) {          // LDS-resident points
            int id = j * FPS_THREADS + t;
            float dx = __fsub_rn(sxyz[id * 3 + 0], px);
            float dy = __fsub_rn(sxyz[id * 3 + 1], py);
            float dz = __fsub_rn(sxyz[id * 3 + 2], pz);
            float dd = __fadd_rn(__fadd_rn(__fmul_rn(dx, dx), __fmul_rn(dy, dy)),
                                 __fmul_rn(dz, dz));
            float nd = fminf(dist[j], dd); dist[j] = nd;
            if (nd > bD) { bD = nd; bI = id; }
        }
#pragma unroll
        for (int j = PPT_LDS; j < PPT; ++j) {        // L2-resident points
            int id = j * FPS_THREADS + t;
            const float* q = gx + (size_t)id * 3;
            float dx = __fsub_rn(q[0], px);
            float dy = __fsub_rn(q[1], py);
            float dz = __fsub_rn(q[2], pz);
            float dd = __fadd_rn(__fadd_rn(__fmul_rn(dx, dx), __fmul_rn(dy, dy)),
                                 __fmul_rn(dz, dz));
            float nd = fminf(dist[j], dd); dist[j] = nd;
            if (nd > bD) { bD = nd; bI = id; }
        }
        // wave32 argmax reduce (first-index wins ties, matching jnp.argmax)
#pragma unroll
        for (int off = 16; off > 0; off >>= 1) {
            float oD = __shfl_xor(bD, off);
            int   oI = __shfl_xor(bI, off);
            if (oD > bD || (oD == bD && oI < bI)) { bD = oD; bI = oI; }
        }
        const int lane = t & 31, wv = t >> 5;
        if (lane == 0) { redD[wv] = bD; redI[wv] = bI; }
        __syncthreads();
        if (wv == 0) {
            bD = redD[lane]; bI = redI[lane];
#pragma unroll
            for (int off = 16; off > 0; off >>= 1) {
                float oD = __shfl_xor(bD, off);
                int   oI = __shfl_xor(bI, off);
                if (oD > bD || (oD == bD && oI < bI)) { bD = oD; bI = oI; }
            }
            if (lane == 0) {
                cidx[b * NGROUP + it] = bI;
                float cx, cy, cz;
                if (bI < LDSP) {
                    cx = sxyz[bI * 3]; cy = sxyz[bI * 3 + 1]; cz = sxyz[bI * 3 + 2];
                } else {
                    const float* q = gx + (size_t)bI * 3;
                    cx = q[0]; cy = q[1]; cz = q[2];
                }
                cPt[0] = cx; cPt[1] = cy; cPt[2] = cz;
                float* co = centerOut + ((size_t)b * NGROUP + it) * 3;
                co[0] = cx; co[1] = cy; co[2] = cz;
            }
        }
        __syncthreads();
    }
}

// ----------------- Kernel 2: random walks (one wave32 per walk) -------------
// Per-step RNG (JAX: split->split->uniform(3)/randint) precomputed across lanes.
__device__ __forceinline__ void walk_step_rng(uint32_t k0, uint32_t k1, int st,
                                              float& u0, float& u1, float& u2,
                                              int& rnd) {
    // kk = split(k, 63)[st]   (counts = arange(126): x0=[0..62], x1=[63..125])
    uint32_t kk0, kk1, j0, j1;
    if (st <= 30) {
        tf2x32(k0, k1, 2u * st,      63u + 2u * st, kk0, j0);
        tf2x32(k0, k1, 2u * st + 1u, 64u + 2u * st, kk1, j1);
    } else if (st == 31) {
        tf2x32(k0, k1, 62u, 125u, kk0, j0);
        tf2x32(k0, k1, 0u,  63u,  j1, kk1);
    } else {
        uint32_t m = 2u * (uint32_t)st - 63u;
        tf2x32(k0, k1, m,      2u * st,      j0, kk0);
        tf2x32(k0, k1, m + 1u, 2u * st + 1u, j1, kk1);
    }
    // ka, kb = split(kk)  -> ka=(A0,A1), kb=(B0,B1)
    uint32_t A0, B0, A1, B1;
    tf2x32(kk0, kk1, 0u, 2u, A0, B0);
    tf2x32(kk0, kk1, 1u, 3u, A1, B1);
    // uniform(ka, (3,)): counts [0,1,2] padded -> bits = [y0[0], y0[1], y1[0]]
    uint32_t b0, b1, b2, jk;
    tf2x32(A0, A1, 0u, 2u, b0, b2);
    tf2x32(A0, A1, 1u, 0u, b1, jk);
    u0 = __uint_as_float((b0 >> 9) | 0x3F800000u) - 1.0f;
    u1 = __uint_as_float((b1 >> 9) | 0x3F800000u) - 1.0f;
    u2 = __uint_as_float((b2 >> 9) | 0x3F800000u) - 1.0f;
    // randint(kb, (), 0, 65536): span=2^16 -> lower word & 0xFFFF
    uint32_t h, l;
    tf2x32(B0, B1, 0u, 1u, h, l);
    rnd = (int)(l & 0xFFFFu);
}

__global__ __launch_bounds__(256) void walk_kernel(
    const int* __restrict__ ring, const int* __restrict__ cidx,
    long long* __restrict__ idxOut) {
    const int wid  = (blockIdx.x * blockDim.x + threadIdx.x) >> 5; // walk id
    const int lane = threadIdx.x & 31;
    const int b = wid >> 9;
    // walk key = split(key(42), 4096)[wid]
    uint32_t k0, k1, d0, d1;
    {
        const uint32_t n = WALKS;           // 4096
        uint32_t i = (uint32_t)wid;
        if (i < n / 2) {
            tf2x32(0u, 42u, 2u * i,      n + 2u * i,      k0, d0);
            tf2x32(0u, 42u, 2u * i + 1u, n + 2u * i + 1u, k1, d1);
        } else {
            uint32_t m = 2u * i - n;
            tf2x32(0u, 42u, m,      2u * i,      d0, k0);
            tf2x32(0u, 42u, m + 1u, 2u * i + 1u, d1, k1);
        }
    }
    // lane L precomputes RNG for steps L and L+32
    float u00 = -1.f, u01 = -1.f, u02 = -1.f;
    float u10 = -1.f, u11 = -1.f, u12 = -1.f;
    int r0s = 0, r1s = 0;
    walk_step_rng(k0, k1, lane, u00, u01, u02, r0s);
    if (lane + 32 < GSIZE - 1)
        walk_step_rng(k0, k1, lane + 32, u10, u11, u12, r1s);

    const int f0 = cidx[wid];
    int v0 = -1, v1 = -1, v2 = -1;       // wave-distributed visited list (96 cap)
    if (lane == 0) v0 = f0;
    int cnt = 1;
    long long* out = idxOut + (size_t)wid * GSIZE;
    if (lane == 0) out[0] = (long long)f0;
    int cur = f0;
    const int* rb = ring + (size_t)b * NPTS * 3;

    for (int st = 0; st < GSIZE - 1; ++st) {
        int n0 = rb[cur * 3 + 0], n1 = rb[cur * 3 + 1], n2 = rb[cur * 3 + 2];
        bool m0 = !__any(v0 == n0 || v1 == n0 || v2 == n0);
        bool m1 = !__any(v0 == n1 || v1 == n1 || v2 == n1);
        bool m2 = !__any(v0 == n2 || v1 == n2 || v2 == n2);
        const int src = st & 31;
        const bool hi = (st >= 32);
        float w0 = __shfl(hi ? u10 : u00, src);
        float w1 = __shfl(hi ? u11 : u01, src);
        float w2 = __shfl(hi ? u12 : u02, src);
        int   rr = __shfl(hi ? r1s : r0s, src);
        float s0 = m0 ? w0 : -1.0f;
        float s1 = m1 ? w1 : -1.0f;
        float s2 = m2 ? w2 : -1.0f;
        int pick = n0; float bs = s0;                 // first-max = jnp.argmax
        if (s1 > bs) { bs = s1; pick = n1; }
        if (s2 > bs) { bs = s2; pick = n2; }
        int nxt = (m0 || m1 || m2) ? pick : rr;
        int owner = cnt & 31, slot = cnt >> 5;        // insert into visited list
        if (lane == owner) {
            if (slot == 0) v0 = nxt; else if (slot == 1) v1 = nxt; else v2 = nxt;
        }
        ++cnt;
        if (lane == 0) out[st + 1] = (long long)nxt;
        cur = nxt;
    }
}

// ----------------------- Kernel 3: gather + recenter ------------------------
__global__ __launch_bounds__(256) void gather_kernel(
    const float* __restrict__ xyz, const long long* __restrict__ idx64,
    const float* __restrict__ centerBuf, float* __restrict__ neigh) {
    int i = blockIdx.x * 256 + threadIdx.x;
    if (i >= B * NGROUP * GSIZE) return;
    int bg = i >> 6;                       // GSIZE = 64
    int b  = bg >> 9;                      // NGROUP = 512
    int id = (int)idx64[i];
    const float* q = xyz + ((size_t)b * NPTS + (size_t)id) * 3;
    const float* c = centerBuf + (size_t)bg * 3;
    float* o = neigh + (size_t)i * 3;
    o[0] = __fsub_rn(q[0], c[0]);
    o[1] = __fsub_rn(q[1], c[1]);
    o[2] = __fsub_rn(q[2], c[2]);
}

// --------------------------------- launch -----------------------------------
extern "C" void kernel_launch(void* const* d_in, const int* in_sizes, int n_in,
                              void* d_out, int out_size, void* d_ws, size_t ws_size,
                              hipStream_t stream) {
    (void)in_sizes; (void)n_in; (void)out_size; (void)ws_size;
    const float* xyz  = (const float*)d_in[0];
    const int*   ring = (const int*)d_in[1];
    // d_in[2]=vertices, d_in[3]=faces: unused by the reference computation.

    const size_t neighElems  = (size_t)B * NGROUP * GSIZE * 3;  // 786432 f32
    const size_t centerElems = (size_t)B * NGROUP * 3;          //  12288 f32
    float*     neighOut  = (float*)d_out;
    float*     centerOut = (float*)d_out + neighElems;
    long long* idxOut    = (long long*)((char*)d_out +
                             (neighElems + centerElems) * sizeof(float));
    int* cidx = (int*)d_ws;                                     // 16 KB scratch

    fps_kernel<<<B, FPS_THREADS, 0, stream>>>(xyz, cidx, centerOut);
    walk_kernel<<<WALKS / 8, 256, 0, stream>>>(ring, cidx, idxOut);
    gather_kernel<<<(B * NGROUP * GSIZE + 255) / 256, 256, 0, stream>>>(
        xyz, idxOut, centerOut, neighOut);
}